// GCN_29978871726566
// MI455X (gfx1250) — compile-verified
//
#include <hip/hip_runtime.h>

typedef __attribute__((ext_vector_type(2))) float v2f;
typedef __attribute__((ext_vector_type(8))) float v8f;

// ---------------------------------------------------------------------------
// Dense GEMM + bias (+optional ReLU on the A operand) using fp32 WMMA.
// One wave computes a 16 x (16*TN) output strip:  C = relu?(A) @ W + bias.
//   A: nrows x K (row major), W: K x M (row major), C: nrows x M.
// V_WMMA_F32_16X16X4_F32 operand layout (ISA 7.12.2):
//   A 16x4 : lanes 0-15 -> M=lane, {K=0,K=1}; lanes 16-31 -> M=lane-16, {K=2,K=3}
//   B 4x16 : lanes 0-15 -> N=lane, rows {K=0,K=1}; lanes 16-31 -> rows {K=2,K=3}
//   C 16x16: VGPR v -> row v (lanes 0-15) / row v+8 (lanes 16-31), col = lane&15
// ---------------------------------------------------------------------------
template <int K, int M, int TN, bool RELU_A>
__global__ __launch_bounds__(256) void gemm_bias_wmma(
    const float* __restrict__ A, const float* __restrict__ W,
    const float* __restrict__ bias, float* __restrict__ C, int nrows)
{
    const int lane = threadIdx.x & 31;
    const int wave = threadIdx.x >> 5;
    const int rt   = blockIdx.x * 8 + wave;          // 16-row tile index
    if (rt * 16 >= nrows) return;                    // wave-uniform guard
    const int colbase = blockIdx.y * (16 * TN);
    const int n  = lane & 15;                        // column within 16-tile
    const int kp = (lane >> 4) << 1;                 // 0 or 2 (K sub-offset)

    // bias-initialized accumulators (column n is identical for both halves)
    v8f acc[TN];
#pragma unroll
    for (int t = 0; t < TN; ++t) {
        const float bv = bias[colbase + t * 16 + n];
#pragma unroll
        for (int v = 0; v < 8; ++v) acc[t][v] = bv;
    }

    const float* __restrict__ arow = A + (size_t)(rt * 16 + n) * K;

#pragma unroll 2
    for (int k = 0; k < K; k += 4) {
        v2f a;
        a.x = arow[k + kp + 0];
        a.y = arow[k + kp + 1];
        if (RELU_A) { a.x = fmaxf(a.x, 0.0f); a.y = fmaxf(a.y, 0.0f); }
#pragma unroll
        for (int t = 0; t < TN; ++t) {
            const float* __restrict__ wp =
                W + (size_t)(k + kp) * M + colbase + t * 16 + n;
            v2f b;
            b.x = wp[0];
            b.y = wp[M];
            acc[t] = __builtin_amdgcn_wmma_f32_16x16x4_f32(
                /*neg_a=*/false, a, /*neg_b=*/false, b,
                /*c_mod=*/(short)0, acc[t],
                /*reuse_a=*/false, /*reuse_b=*/false);
        }
    }

    const int mhalf = (lane >> 4) * 8;
#pragma unroll
    for (int t = 0; t < TN; ++t)
#pragma unroll
        for (int v = 0; v < 8; ++v)
            C[(size_t)(rt * 16 + mhalf + v) * M + colbase + t * 16 + n] = acc[t][v];
}

// ---------------------------------------------------------------------------
// COO SpMM scatter: out[row[e], :] += vals[e] * h[col[e], :]
// One wave per edge; lanes stream float4 chunks of the D-wide feature row.
// h panels (<=102 MB) are L2-resident; atomics resolve in L2 atomic units.
// ---------------------------------------------------------------------------
__global__ __launch_bounds__(256) void spmm_edge_atomic(
    const int* __restrict__ row, const int* __restrict__ col,
    const float* __restrict__ vals, const float* __restrict__ h,
    float* __restrict__ out, int nedges, int D)
{
    const long gid  = (long)blockIdx.x * blockDim.x + threadIdx.x;
    const int  e    = (int)(gid >> 5);
    const int  lane = (int)(gid & 31);
    if (e >= nedges) return;                         // wave-uniform guard
    const int   r = row[e];
    const int   c = col[e];
    const float v = vals[e];
    const float4* __restrict__ hp = (const float4*)(h + (size_t)c * D);
    float* __restrict__ op = out + (size_t)r * D;
    for (int q = lane; q * 4 < D; q += 32) {
        const float4 hv = hp[q];
        unsafeAtomicAdd(op + q * 4 + 0, v * hv.x);
        unsafeAtomicAdd(op + q * 4 + 1, v * hv.y);
        unsafeAtomicAdd(op + q * 4 + 2, v * hv.z);
        unsafeAtomicAdd(op + q * 4 + 3, v * hv.w);
    }
}

__global__ __launch_bounds__(256) void zero_f4(float4* __restrict__ p, long n4)
{
    const long i = (long)blockIdx.x * blockDim.x + threadIdx.x;
    if (i < n4) p[i] = make_float4(0.f, 0.f, 0.f, 0.f);
}

// ---------------------------------------------------------------------------
extern "C" void kernel_launch(void* const* d_in, const int* in_sizes, int n_in,
                              void* d_out, int out_size, void* d_ws, size_t ws_size,
                              hipStream_t stream)
{
    (void)in_sizes; (void)n_in; (void)out_size; (void)ws_size;

    const float* x    = (const float*)d_in[0];
    const int*   row  = (const int*)  d_in[1];
    const int*   col  = (const int*)  d_in[2];
    const float* vals = (const float*)d_in[3];
    const float* W1   = (const float*)d_in[4];
    const float* b1   = (const float*)d_in[5];
    const float* W2   = (const float*)d_in[6];
    const float* b2   = (const float*)d_in[7];
    const float* W3   = (const float*)d_in[8];
    const float* b3   = (const float*)d_in[9];
    float* out = (float*)d_out;

    constexpr int N = 100000;
    constexpr int E = 1600000;

    float* B0 = (float*)d_ws;                 // N x 256 panel (GEMM outputs)
    float* B1 = B0 + (size_t)N * 256;         // N x 256 panel (SpMM outputs)

    const dim3 blk(256);
    const dim3 gGemm256((N / 16 + 7) / 8, 2); // TN=8 -> 128 cols/wave, 2 groups
    const dim3 gGemm64 ((N / 16 + 7) / 8, 1); // TN=4 -> 64 cols/wave
    const int  gSpmm   = (E * 32) / 256;      // wave per edge
    const int  gZero256 = ((long)N * 256 / 4 + 255) / 256;
    const int  gZero64  = ((long)N * 64 / 4 + 255) / 256;

    // ---- Layer 1: Hpre = X @ W1 + b1 ; h1 = relu(spmm(Hpre)) (relu deferred)
    gemm_bias_wmma<512, 256, 8, false><<<gGemm256, blk, 0, stream>>>(x, W1, b1, B0, N);
    zero_f4<<<gZero256, blk, 0, stream>>>((float4*)B1, (long)N * 256 / 4);
    spmm_edge_atomic<<<gSpmm, blk, 0, stream>>>(row, col, vals, B0, B1, E, 256);

    // ---- Layer 2: Hpre = relu(h1) @ W2 + b2 ; h2 = relu(spmm(Hpre)) (deferred)
    gemm_bias_wmma<256, 256, 8, true><<<gGemm256, blk, 0, stream>>>(B1, W2, b2, B0, N);
    zero_f4<<<gZero256, blk, 0, stream>>>((float4*)B1, (long)N * 256 / 4);
    spmm_edge_atomic<<<gSpmm, blk, 0, stream>>>(row, col, vals, B0, B1, E, 256);

    // ---- Layer 3: Hpre = relu(h2) @ W3 + b3 ; out = spmm(Hpre)  (no relu)
    gemm_bias_wmma<256, 64, 4, true><<<gGemm64, blk, 0, stream>>>(B1, W3, b3, B0, N);
    zero_f4<<<gZero64, blk, 0, stream>>>((float4*)out, (long)N * 64 / 4);
    spmm_edge_atomic<<<gSpmm, blk, 0, stream>>>(row, col, vals, B0, out, E, 64);
}